// GIN_56659208568912
// MI455X (gfx1250) — compile-verified
//
#include <hip/hip_runtime.h>

typedef __attribute__((ext_vector_type(2))) float v2f;
typedef __attribute__((ext_vector_type(8))) float v8f;

#define HDIM 128

// ---------------- elementwise helpers ----------------

__global__ void copy_f4(const float4* __restrict__ in, float4* __restrict__ out, long n) {
  long i = (long)blockIdx.x * blockDim.x + threadIdx.x;
  if (i < n) out[i] = in[i];
}

__global__ void zero_f(float* __restrict__ p, int n) {
  int i = blockIdx.x * blockDim.x + threadIdx.x;
  if (i < n) p[i] = 0.f;
}

// ---------------- edge scatter-add: agg[dst] += h[src] ----------------
// one lane per (edge, 4-feature group); float4 load + 4 f32 atomics
__global__ void scatter_add(const float* __restrict__ h,
                            const long long* __restrict__ src,
                            const long long* __restrict__ dst,
                            float* __restrict__ agg, int E) {
  long i = (long)blockIdx.x * blockDim.x + threadIdx.x;
  int e = (int)(i >> 5);
  int q = (int)(i & 31);
  if (e >= E) return;
  long long s = src[e], d = dst[e];
  const float4 v = *(const float4*)(h + (size_t)s * HDIM + q * 4);
  float* p = agg + (size_t)d * HDIM + q * 4;
  atomicAdd(p + 0, v.x);
  atomicAdd(p + 1, v.y);
  atomicAdd(p + 2, v.z);
  atomicAdd(p + 3, v.w);
}

// ---------------- f32 WMMA GEMM: C[nrows x NCOLS] = act(sum_s A_s @ W_s + b) ----
// A segments: A + s*segStride, each [nrows x 128] row-major (stride 128).
// W: [nsegs*128 x NCOLS] row-major. Block = 256 threads = 8 waves; wave w owns
// rows [blk*128 + 16w, +16).
// W segment staged in LDS in B-fragment pair order:
//   sW[4*(g*NCOLS + n) + 2h + v] = W[4g + 2h + v][n],  g = k/4, h = lane/16, v = 0/1
// so each B fragment is ONE contiguous ds_load_b64 into an aligned VGPR pair
// (conflict-free: half-wave 0 hits dword banks {4n,4n+1}, half-wave 1 {4n+2,4n+3}).
template <int NCOLS>
__launch_bounds__(256)
__global__ void gemm_wmma(const float* __restrict__ Abase, long segStride,
                          const float* __restrict__ W,
                          const float* __restrict__ bias, float* __restrict__ C,
                          int nrows, int nsegs, int relu) {
  constexpr int NT = NCOLS / 16;
  __shared__ __align__(16) float sW[128 * NCOLS];  // 64KB @ NCOLS=128

  const int tid = threadIdx.x;
  const int wave = tid >> 5;
  const int lane = tid & 31;
  const int lhalf = lane >> 4;   // 0/1: K-pair select (A/B), M-half (C)
  const int lmod = lane & 15;
  const int rowbase = blockIdx.x * 128 + wave * 16;
  const bool active = rowbase < nrows;   // nrows % 16 == 0 -> strips all-or-nothing
  const int row = rowbase + lmod;

  v8f acc[NT] = {};

  for (int s = 0; s < nsegs; ++s) {
    __syncthreads();
    const float* Wseg = W + (size_t)s * 128 * NCOLS;
    // stage W into fragment-pair-interleaved LDS layout (coalesced global reads)
    for (int p = tid; p < 64 * NCOLS; p += 256) {
      const int j = p / NCOLS;       // K-pair index 0..63
      const int n = p % NCOLS;
      const int g = j >> 1, h = j & 1;
      const float w0 = Wseg[(2 * j + 0) * NCOLS + n];
      const float w1 = Wseg[(2 * j + 1) * NCOLS + n];
      *(float2*)(sW + 4 * (g * NCOLS + n) + 2 * h) = make_float2(w0, w1);
    }
    __syncthreads();
    if (!active) continue;  // barriers above are hit by every thread each iter

    const float* A = Abase + (size_t)s * segStride + (size_t)row * HDIM;
    for (int k = 0; k < 128; k += 4) {
      // A frag (16x4 f32): lane holds A[row][k + 2*lhalf + {0,1}] -> one b64 load
      float2 av = *(const float2*)(A + k + lhalf * 2);
      v2f a; a.x = av.x; a.y = av.y;
      const float* bbase = sW + (k >> 2) * (4 * NCOLS) + 4 * lmod + 2 * lhalf;
#pragma unroll
      for (int t = 0; t < NT; ++t) {
        // B frag (4x16 f32): one contiguous ds_load_b64, immediate offset 64*t dwords
        v2f b = *(const v2f*)(bbase + 64 * t);
        acc[t] = __builtin_amdgcn_wmma_f32_16x16x4_f32(
            false, a, false, b, (short)0, acc[t], false, false);
      }
    }
  }
  if (!active) return;

  // C frag layout: vgpr i, lanes 0-15 -> (M=i, N=lane); lanes 16-31 -> (M=i+8)
#pragma unroll
  for (int t = 0; t < NT; ++t) {
    const int n = t * 16 + lmod;
    const float bv = bias[n];
#pragma unroll
    for (int i = 0; i < 8; ++i) {
      const int r = rowbase + i + 8 * lhalf;
      float v = acc[t][i] + bv;
      if (relu) v = fmaxf(v, 0.f);
      C[(size_t)r * NCOLS + n] = v;
    }
  }
}

// ---------------- BatchNorm (training-mode, over nodes) ----------------

__global__ void bn_stats(const float* __restrict__ z, float* __restrict__ stats, int nrows) {
  // blockDim = 128 (one column per thread); each block covers 512 rows
  const int c = threadIdx.x;
  int r0 = blockIdx.x * 512;
  int r1 = r0 + 512; if (r1 > nrows) r1 = nrows;
  float s = 0.f, s2 = 0.f;
  for (int r = r0; r < r1; ++r) {
    float v = z[(size_t)r * HDIM + c];
    s += v; s2 += v * v;
  }
  atomicAdd(&stats[c], s);
  atomicAdd(&stats[HDIM + c], s2);
}

__global__ void bn_apply(const float* __restrict__ z, const float* __restrict__ stats,
                         const float* __restrict__ gamma, const float* __restrict__ beta,
                         float* __restrict__ out, int nrows, int relu) {
  long i = (long)blockIdx.x * blockDim.x + threadIdx.x;  // float4 index
  long total = (long)nrows * (HDIM / 4);
  if (i >= total) return;
  const int cq = (int)(i & (HDIM / 4 - 1));
  const float inv = 1.f / (float)nrows;
  float4 v = ((const float4*)z)[i];
  float4 o;
  const float* vi = (const float*)&v;
  float* vo = (float*)&o;
#pragma unroll
  for (int j = 0; j < 4; ++j) {
    const int c = cq * 4 + j;
    const float mean = stats[c] * inv;
    const float var = stats[HDIM + c] * inv - mean * mean;
    float r = (vi[j] - mean) * rsqrtf(var + 1e-5f) * gamma[c] + beta[c];
    if (relu) r = fmaxf(r, 0.f);
    vo[j] = r;
  }
  ((float4*)out)[i] = o;
}

// ---------------- driver ----------------

extern "C" void kernel_launch(void* const* d_in, const int* in_sizes, int n_in,
                              void* d_out, int out_size, void* d_ws, size_t ws_size,
                              hipStream_t stream) {
  const float* x         = (const float*)d_in[0];
  const long long* ei    = (const long long*)d_in[1];  // int64 [2, E]
  const float* convW     = (const float*)d_in[2];      // [3,3,128,128]
  const float* convB     = (const float*)d_in[3];      // [3,3,128]
  const float* gamma     = (const float*)d_in[4];      // [3,128]
  const float* beta      = (const float*)d_in[5];      // [3,128]
  const float* linW      = (const float*)d_in[6];      // [384,128]
  const float* linB      = (const float*)d_in[7];      // [128]
  const float* clsW      = (const float*)d_in[8];      // [128,64]
  const float* clsB      = (const float*)d_in[9];      // [64]

  const int N = in_sizes[0] / HDIM;  // 50000
  const int E = in_sizes[1] / 2;     // 800000
  const long long* src = ei;
  const long long* dst = ei + E;

  float* ws = (float*)d_ws;
  const size_t NH = (size_t)N * HDIM;
  float* bufZ  = ws;            // z / mlp2 scratch
  float* bufT  = ws + NH;       // mlp1 / z3 scratch
  float* h1    = ws + 2 * NH;   // h1,h2,h3 contiguous -> concat GEMM segStride = NH
  float* h2    = ws + 3 * NH;
  float* h3    = ws + 4 * NH;
  float* stats = ws + 5 * NH;   // [2][128] sum / sumsq
  float* hl[3] = {h1, h2, h3};

  float* logits = (float*)d_out;                    // [N,64]
  float* hout   = (float*)d_out + (size_t)N * 64;   // [N,128]

  const dim3 blk(256);
  const int gemmBlocks  = (N + 127) / 128;
  const long copyN      = (long)(NH / 4);
  const int copyBlocks  = (int)((copyN + 255) / 256);
  const long scatN      = (long)E * 32;
  const int scatBlocks  = (int)((scatN + 255) / 256);
  const int bnBlocks    = (N + 511) / 512;
  const int applyBlocks = copyBlocks;

  const float* hin = x;
  for (int l = 0; l < 3; ++l) {
    // z = h + segment_sum(h[src] -> dst)
    copy_f4<<<copyBlocks, blk, 0, stream>>>((const float4*)hin, (float4*)bufZ, copyN);
    scatter_add<<<scatBlocks, blk, 0, stream>>>(hin, src, dst, bufZ, E);

    const float* W0 = convW + (size_t)(l * 3 + 0) * HDIM * HDIM;
    const float* W1 = convW + (size_t)(l * 3 + 1) * HDIM * HDIM;
    const float* W2 = convW + (size_t)(l * 3 + 2) * HDIM * HDIM;
    const float* b0 = convB + (l * 3 + 0) * HDIM;
    const float* b1 = convB + (l * 3 + 1) * HDIM;
    const float* b2 = convB + (l * 3 + 2) * HDIM;

    gemm_wmma<128><<<gemmBlocks, blk, 0, stream>>>(bufZ, 0, W0, b0, bufT, N, 1, 1);
    gemm_wmma<128><<<gemmBlocks, blk, 0, stream>>>(bufT, 0, W1, b1, bufZ, N, 1, 1);
    gemm_wmma<128><<<gemmBlocks, blk, 0, stream>>>(bufZ, 0, W2, b2, bufT, N, 1, 0);

    zero_f<<<1, 256, 0, stream>>>(stats, 2 * HDIM);
    bn_stats<<<bnBlocks, dim3(128), 0, stream>>>(bufT, stats, N);
    bn_apply<<<applyBlocks, blk, 0, stream>>>(bufT, stats, gamma + l * HDIM, beta + l * HDIM,
                                              hl[l], N, (l < 2) ? 1 : 0);
    hin = hl[l];
  }

  // h = concat(h1,h2,h3) @ linW + linB   (3 contiguous K-segments of 128)
  gemm_wmma<128><<<gemmBlocks, blk, 0, stream>>>(h1, (long)NH, linW, linB, hout, N, 3, 0);
  // logits = h @ clsW + clsB
  gemm_wmma<64><<<gemmBlocks, blk, 0, stream>>>(hout, 0, clsW, clsB, logits, N, 1, 0);
}